// GNN_75814762709760
// MI455X (gfx1250) — compile-verified
//
#include <hip/hip_runtime.h>
#include <math.h>

typedef float v2f __attribute__((ext_vector_type(2)));
typedef float v8f __attribute__((ext_vector_type(8)));

#define N_NODES 50000
#define DIM     128
#define N_EDGES 800000

// ---------------------------------------------------------------------------
// Degree / normalization
// ---------------------------------------------------------------------------
__global__ void deg_init_kernel(float* __restrict__ deg) {
    int i = blockIdx.x * 256 + threadIdx.x;
    if (i < N_NODES) deg[i] = 1.0f;  // self-loop contributes 1 to every node
}

__global__ void deg_count_kernel(const int* __restrict__ dst,
                                 float* __restrict__ deg) {
    int e = blockIdx.x * 256 + threadIdx.x;
    if (e < N_EDGES) atomicAdd(&deg[dst[e]], 1.0f);
}

__global__ void deg_rsqrt_kernel(float* __restrict__ deg) {
    int i = blockIdx.x * 256 + threadIdx.x;
    if (i < N_NODES) {
        float d = deg[i];               // d >= 1 always
        deg[i] = rsqrtf(d);
    }
}

// ---------------------------------------------------------------------------
// GEMM: C[50000,128] = A[50000,128] x B[128,128]  via V_WMMA_F32_16X16X4_F32
// One wave computes one 16x16 tile of C. Block = 256 threads = 8 waves = the
// 8 N-tiles of one 16-row M stripe. Grid = 3125 blocks (50000/16 exactly).
// No divergence anywhere: EXEC all-1s as WMMA requires.
// ---------------------------------------------------------------------------
__global__ __launch_bounds__(256) void gemm_wmma_kernel(
        const float* __restrict__ A, const float* __restrict__ B,
        float* __restrict__ C) {
    const int lane = threadIdx.x & 31;
    const int nt   = threadIdx.x >> 5;   // N tile 0..7
    const int mt   = blockIdx.x;         // M tile 0..3124
    const int half = lane >> 4;          // which 16-lane half
    const int l16  = lane & 15;

    // A fragment: lane (m = l16), VGPR j holds A[m][k + 2*half + j]
    const float* a_row = A + (mt * 16 + l16) * DIM;
    // B fragment: lane (n = l16), VGPR j holds B[k + 2*half + j][n]
    const float* b_col = B + nt * 16 + l16;

    v8f acc = {};
    #pragma unroll
    for (int k = 0; k < DIM; k += 4) {
        int ka = k + 2 * half;
        v2f a, b;
        a.x = a_row[ka];
        a.y = a_row[ka + 1];
        b.x = b_col[ka * DIM];
        b.y = b_col[(ka + 1) * DIM];
        acc = __builtin_amdgcn_wmma_f32_16x16x4_f32(
            /*neg_a=*/false, a, /*neg_b=*/false, b,
            /*c_mod=*/(short)0, acc, /*reuse_a=*/false, /*reuse_b=*/false);
    }

    // C/D layout: VGPR v -> M = v + 8*half, N = l16
    float* c_base = C + nt * 16 + l16;
    const int rbase = mt * 16 + 8 * half;
    #pragma unroll
    for (int v = 0; v < 8; ++v)
        c_base[(rbase + v) * DIM] = acc[v];
}

// ---------------------------------------------------------------------------
// Self-loop init: out[i,:] = h[i,:] * dinv[i]^2   (float4 per lane)
// ---------------------------------------------------------------------------
__global__ void agg_init_kernel(const float* __restrict__ h,
                                const float* __restrict__ dinv,
                                float* __restrict__ out) {
    int idx  = blockIdx.x * 256 + threadIdx.x;   // N_NODES*32 work items
    int node = idx >> 5;
    int l    = idx & 31;
    if (node >= N_NODES) return;
    float s = dinv[node];
    s = s * s;
    float4 v = ((const float4*)(h + node * DIM))[l];
    float4 o;
    o.x = v.x * s; o.y = v.y * s; o.z = v.z * s; o.w = v.w * s;
    ((float4*)(out + node * DIM))[l] = o;
}

// ---------------------------------------------------------------------------
// Edge scatter: out[dst,:] += h[src,:] * dinv[src]*dinv[dst]
// One wave per edge: 32 lanes x float4 = 128 features. fp32 global atomics
// (no-return -> STOREcnt path), mostly L2-resident (~110MB working set
// inside the 192MB L2).
// ---------------------------------------------------------------------------
__global__ __launch_bounds__(256) void edge_scatter_kernel(
        const int* __restrict__ src, const int* __restrict__ dst,
        const float* __restrict__ h, const float* __restrict__ dinv,
        float* __restrict__ out) {
    int lane = threadIdx.x & 31;
    int e    = blockIdx.x * 8 + (threadIdx.x >> 5);
    if (e >= N_EDGES) return;
    int s = src[e];
    int d = dst[e];
    float norm = dinv[s] * dinv[d];
    float4 v = ((const float4*)(h + (long)s * DIM))[lane];
    float* op = out + (long)d * DIM + lane * 4;
    atomicAdd(op + 0, v.x * norm);
    atomicAdd(op + 1, v.y * norm);
    atomicAdd(op + 2, v.z * norm);
    atomicAdd(op + 3, v.w * norm);
}

// ---------------------------------------------------------------------------
// Epilogues
// ---------------------------------------------------------------------------
__global__ void bias_relu_kernel(float* __restrict__ x,
                                 const float* __restrict__ b) {
    int i = blockIdx.x * 256 + threadIdx.x;
    if (i < N_NODES * DIM) {
        float v = x[i] + b[i & (DIM - 1)];
        x[i] = v > 0.0f ? v : 0.0f;
    }
}

__global__ __launch_bounds__(256) void bias_logsoftmax_kernel(
        float* __restrict__ x, const float* __restrict__ b) {
    int lane = threadIdx.x & 31;
    int node = blockIdx.x * 8 + (threadIdx.x >> 5);
    if (node >= N_NODES) return;
    float4 v  = ((const float4*)(x + (long)node * DIM))[lane];
    float4 bb = ((const float4*)b)[lane];
    v.x += bb.x; v.y += bb.y; v.z += bb.z; v.w += bb.w;

    float m = fmaxf(fmaxf(v.x, v.y), fmaxf(v.z, v.w));
    #pragma unroll
    for (int off = 16; off >= 1; off >>= 1)
        m = fmaxf(m, __shfl_xor(m, off, 32));

    float s = expf(v.x - m) + expf(v.y - m) + expf(v.z - m) + expf(v.w - m);
    #pragma unroll
    for (int off = 16; off >= 1; off >>= 1)
        s += __shfl_xor(s, off, 32);

    float lse = m + logf(s);
    v.x -= lse; v.y -= lse; v.z -= lse; v.w -= lse;
    ((float4*)(x + (long)node * DIM))[lane] = v;
}

// ---------------------------------------------------------------------------
// Launch
// ---------------------------------------------------------------------------
extern "C" void kernel_launch(void* const* d_in, const int* in_sizes, int n_in,
                              void* d_out, int out_size, void* d_ws, size_t ws_size,
                              hipStream_t stream) {
    const float* x  = (const float*)d_in[0];
    const int*   ei = (const int*)  d_in[1];   // [2, E] row-major
    const float* W1 = (const float*)d_in[2];
    const float* b1 = (const float*)d_in[3];
    const float* W2 = (const float*)d_in[4];
    const float* b2 = (const float*)d_in[5];
    float* out = (float*)d_out;

    float* ws   = (float*)d_ws;
    float* dinv = ws;                             //  50000 floats
    float* h    = ws + N_NODES;                   //  6.4M floats
    float* a    = h + (size_t)N_NODES * DIM;      //  6.4M floats

    const int* src = ei;
    const int* dst = ei + N_EDGES;

    const int nblk_nodes = (N_NODES + 255) / 256;            // 196
    const int nblk_edges = (N_EDGES + 255) / 256;            // 3125
    const int nblk_gemm  = N_NODES / 16;                     // 3125
    const int nblk_rows  = (N_NODES + 7) / 8;                // 6250 (wave-per-row)
    const int nblk_escat = (N_EDGES + 7) / 8;                // 100000
    const int nblk_elem  = (N_NODES * DIM + 255) / 256;      // 25000
    const int nblk_init  = (N_NODES * 32 + 255) / 256;       // 6250

    // degree -> dinv (self-loop folded into init = 1)
    deg_init_kernel <<<nblk_nodes, 256, 0, stream>>>(dinv);
    deg_count_kernel<<<nblk_edges, 256, 0, stream>>>(dst, dinv);
    deg_rsqrt_kernel<<<nblk_nodes, 256, 0, stream>>>(dinv);

    // Layer 1
    gemm_wmma_kernel   <<<nblk_gemm,  256, 0, stream>>>(x, W1, h);
    agg_init_kernel    <<<nblk_init,  256, 0, stream>>>(h, dinv, a);
    edge_scatter_kernel<<<nblk_escat, 256, 0, stream>>>(src, dst, h, dinv, a);
    bias_relu_kernel   <<<nblk_elem,  256, 0, stream>>>(a, b1);

    // Layer 2
    gemm_wmma_kernel   <<<nblk_gemm,  256, 0, stream>>>(a, W2, h);
    agg_init_kernel    <<<nblk_init,  256, 0, stream>>>(h, dinv, out);
    edge_scatter_kernel<<<nblk_escat, 256, 0, stream>>>(src, dst, h, dinv, out);
    bias_logsoftmax_kernel<<<nblk_rows, 256, 0, stream>>>(out, b2);
}